// CompRGCNEncoder_50723563765985
// MI455X (gfx1250) — compile-verified
//
#include <hip/hip_runtime.h>

// ---------------------------------------------------------------------------
// CompGCN encoder on MI455X (gfx1250, wave32).
// ccorr(a,b)@W == a @ W'_b with W'_b[j][o] = sum_k b[(j+k)%d] W[k][o].
// Edges counting-sorted by relation type into 64-aligned padded chunks so
// messages become dense 16x16 WMMA tiles (V_WMMA_F32_16X16X4_F32).
// LDS staging uses the CDNA5 async global->LDS path (ASYNCcnt).
// ---------------------------------------------------------------------------

typedef float v2f __attribute__((ext_vector_type(2)));
typedef float v8f __attribute__((ext_vector_type(8)));

constexpr int NUM_ENT = 100000;
constexpr int E_HALF  = 1600000;          // edges per direction
constexpr int E_TOT   = 3200000;          // 2*E_HALF
constexpr int NREL2   = 100;              // 2*NUM_REL relation types on edges
constexpr int CHUNK   = 64;               // sort padding granularity
constexpr int NCHUNK  = E_HALF / CHUNK + NREL2;   // 25100 worst case
constexpr int NCHPAD  = NCHUNK * CHUNK;           // padded edge slots / dir

// ----------------------- async global->LDS staging -------------------------
// CDNA5 GLOBAL_LOAD_ASYNC_TO_LDS_B128: per-lane 16B DMA into LDS, no VGPR
// round-trip, tracked by ASYNCcnt. LDS byte offset = low 32 bits of the
// generic pointer (LDS aperture maps by truncation).
__device__ __forceinline__ void async_ld_b128(const float* g, float* l) {
  unsigned ldsoff = (unsigned)(unsigned long long)l;
  asm volatile("global_load_async_to_lds_b128 %0, %1, off"
               :: "v"(ldsoff), "v"(g) : "memory");
}
__device__ __forceinline__ void async_wait0() {
  asm volatile("s_wait_asynccnt 0x0" ::: "memory");
}

// ---------------------------- structural kernels ---------------------------

__global__ void deg_kernel(const int* __restrict__ ei,
                           float* __restrict__ deg_in, float* __restrict__ deg_out) {
  for (int e = blockIdx.x * blockDim.x + threadIdx.x; e < E_HALF;
       e += gridDim.x * blockDim.x) {
    atomicAdd(&deg_in [ei[e]],          1.0f);   // rows of 'in'  direction
    atomicAdd(&deg_out[ei[E_HALF + e]], 1.0f);   // rows of 'out' direction
  }
}

__global__ void dinv_kernel(float* __restrict__ a, float* __restrict__ b) {
  int i = blockIdx.x * blockDim.x + threadIdx.x;
  if (i >= NUM_ENT) return;
  float da = a[i], db = b[i];
  a[i] = (da > 0.f) ? rsqrtf(da) : 0.f;
  b[i] = (db > 0.f) ? rsqrtf(db) : 0.f;
}

__global__ void hist_kernel(const int* __restrict__ etype, int* __restrict__ cnt) {
  __shared__ int h[256];
  int tid = threadIdx.x;
  h[tid] = 0;
  __syncthreads();
  for (int e = blockIdx.x * blockDim.x + tid; e < E_TOT;
       e += gridDim.x * blockDim.x) {
    int dir = (e < E_HALF) ? 0 : 1;
    atomicAdd(&h[dir * 128 + etype[e]], 1);
  }
  __syncthreads();
  if (h[tid]) atomicAdd(&cnt[tid], h[tid]);
}

// single thread: compute 64-aligned segment starts per (dir, type)
__global__ void plan_kernel(const int* __restrict__ cnt,
                            int* __restrict__ seg_fill,      // [2*128] edge cursors
                            int* __restrict__ segcs) {       // [2*101] chunk starts
  for (int dir = 0; dir < 2; ++dir) {
    int chunk = 0;
    for (int t = 0; t < NREL2; ++t) {
      segcs[dir * 101 + t]    = chunk;
      seg_fill[dir * 128 + t] = chunk * CHUNK;
      chunk += (cnt[dir * 128 + t] + CHUNK - 1) / CHUNK;
    }
    segcs[dir * 101 + NREL2] = chunk;     // total used chunks
  }
}

__global__ void chunkfill_kernel(const int* __restrict__ segcs,
                                 int* __restrict__ ctype) {
  int tid = blockIdx.x * blockDim.x + threadIdx.x;
  if (tid >= 2 * NCHUNK) return;
  int dir = tid / NCHUNK, c = tid % NCHUNK;
  const int* s = segcs + dir * 101;
  int out = -1;
  if (c < s[NREL2]) {                     // largest t with s[t] <= c
    int lo = 0, hi = NREL2 - 1;
    while (lo < hi) { int mid = (lo + hi + 1) >> 1; if (s[mid] <= c) lo = mid; else hi = mid - 1; }
    out = lo;
  }
  ctype[dir * NCHUNK + c] = out;
}

__global__ void scatter_kernel(const int* __restrict__ etype,
                               int* __restrict__ seg_fill,
                               int* __restrict__ sorted) {   // [2*NCHPAD], pre-set to -1
  for (int e = blockIdx.x * blockDim.x + threadIdx.x; e < E_TOT;
       e += gridDim.x * blockDim.x) {
    int dir = (e < E_HALF) ? 0 : 1;
    int le  = dir ? (e - E_HALF) : e;
    int pos = atomicAdd(&seg_fill[dir * 128 + etype[e]], 1);
    sorted[dir * NCHPAD + pos] = le;
  }
}

// ------------------------- relation-matrix precompute -----------------------
// Wp[t][j][o] = sum_k rel[t][(j+k)%DIN] * W[k][o]
__global__ void wprime_kernel(const float* __restrict__ rel, const float* __restrict__ W,
                              float* __restrict__ Wp, int DIN, int DOUT) {
  int j = blockIdx.x, t = blockIdx.y, o = threadIdx.x;
  __shared__ float r[128];
  for (int k = o; k < DIN; k += blockDim.x) r[k] = rel[t * DIN + k];
  __syncthreads();
  float s = 0.f;
  for (int k = 0; k < DIN; ++k) {
    int idx = j + k; if (idx >= DIN) idx -= DIN;
    s += r[idx] * W[k * DOUT + o];
  }
  Wp[((size_t)t * DIN + j) * DOUT + o] = s;
}

// ----------------------------- WMMA GEMM cores ------------------------------
// Layout assumptions (ISA 7.12.2, wave32):
//  A 16x4 f32 : lane l (l<16) row M=l, K={0,1}; lanes 16-31 K={2,3}
//  B 4x16 f32 : vgpr v, half h -> K = v + 2h, N = lane&15
//  C/D 16x16  : vgpr v, half h -> M = v + 8h, N = lane&15

template<int DIN, int DOUT, int EPB>      // EPB edges per block (16-multiple)
__global__ void __launch_bounds__(128)
edge_wmma_kernel(const float* __restrict__ x,        // [NUM_ENT][DIN]
                 const float* __restrict__ Wp,       // [NREL2][DIN][DOUT]
                 const int*   __restrict__ sorted,   // padded sorted edge ids
                 const int*   __restrict__ ctype,    // chunk -> type (-1 unused)
                 const int*   __restrict__ rows,
                 const int*   __restrict__ cols,
                 const float* __restrict__ dinv,
                 float*       __restrict__ acc) {    // [NUM_ENT][DOUT]
  constexpr int BPC = CHUNK / EPB;                   // blocks per sorted chunk
  int chunk = blockIdx.x / BPC, sub = blockIdx.x % BPC;
  int t = ctype[chunk];
  if (t < 0) return;

  __shared__ float Xs[EPB * DIN];
  __shared__ float Ws[DIN * DOUT];
  __shared__ int   erow[EPB];
  __shared__ int   ecol[EPB];
  __shared__ float enorm[EPB];

  int tid = threadIdx.x;
  int ebase = chunk * CHUNK + sub * EPB;
  if (tid < EPB) {
    int eid = sorted[ebase + tid];
    int rr = 0, cc = 0; float nv = 0.f;
    if (eid >= 0) { rr = rows[eid]; cc = cols[eid]; nv = dinv[rr] * dinv[cc]; }
    erow[tid] = rr; ecol[tid] = cc; enorm[tid] = nv;
  }
  __syncthreads();

  // async-DMA stage: X rows (L2-resident gather) and W'_t, 16B per lane/issue
  #pragma unroll
  for (int it = 0; it < (EPB * DIN / 4) / 128; ++it) {
    int lin = it * 128 + tid;
    int j = lin / (DIN / 4), kq = lin % (DIN / 4);
    async_ld_b128(x + (size_t)ecol[j] * DIN + kq * 4, &Xs[j * DIN + kq * 4]);
  }
  const float* wsrc = Wp + (size_t)t * DIN * DOUT;
  #pragma unroll
  for (int it = 0; it < (DIN * DOUT / 4) / 128; ++it) {
    int lin = it * 128 + tid;
    async_ld_b128(wsrc + lin * 4, &Ws[lin * 4]);
  }
  async_wait0();
  __syncthreads();

  int wave = tid >> 5, lane = tid & 31;
  int h = lane >> 4, l16 = lane & 15;
  constexpr int NRT = EPB / 16;                      // row tiles in this block
  int rt = wave % NRT;
  const float* xp = &Xs[(rt * 16 + l16) * DIN + 2 * h];
  for (int ct = wave / NRT; ct < DOUT / 16; ct += 4 / NRT) {
    int n0 = ct * 16;
    const float* wpp = &Ws[2 * h * DOUT + n0 + l16];
    v8f c = {};
    #pragma unroll
    for (int s = 0; s < DIN / 4; ++s) {
      v2f a; a.x = xp[4 * s];           a.y = xp[4 * s + 1];
      v2f b; b.x = wpp[4 * s * DOUT];   b.y = wpp[(4 * s + 1) * DOUT];
      c = __builtin_amdgcn_wmma_f32_16x16x4_f32(false, a, false, b, (short)0, c,
                                                false, false);
    }
    // Branchless scatter: padding edges have norm==0 and erow==0, so their
    // contribution is +0.0f -- cheaper than 8 exec-mask branches per tile.
    #pragma unroll
    for (int r = 0; r < 8; ++r) {
      int j = rt * 16 + r + 8 * h;
      atomicAdd(&acc[(size_t)erow[j] * DOUT + n0 + l16], c[r] * enorm[j]);
    }
  }
}

template<int DIN, int DOUT, int RPB>
__global__ void __launch_bounds__(128)
loop_wmma_kernel(const float* __restrict__ x, const float* __restrict__ Wploop,
                 float* __restrict__ acc) {
  int base = blockIdx.x * RPB;
  __shared__ float Xs[RPB * DIN];
  __shared__ float Ws[DIN * DOUT];
  int tid = threadIdx.x;
  #pragma unroll
  for (int it = 0; it < (RPB * DIN / 4) / 128; ++it) {
    int lin = it * 128 + tid;
    int j = lin / (DIN / 4), kq = lin % (DIN / 4);
    int row = base + j; if (row >= NUM_ENT) row = NUM_ENT - 1;   // tail clamp
    async_ld_b128(x + (size_t)row * DIN + kq * 4, &Xs[j * DIN + kq * 4]);
  }
  #pragma unroll
  for (int it = 0; it < (DIN * DOUT / 4) / 128; ++it) {
    int lin = it * 128 + tid;
    async_ld_b128(Wploop + lin * 4, &Ws[lin * 4]);
  }
  async_wait0();
  __syncthreads();

  int wave = tid >> 5, lane = tid & 31;
  int h = lane >> 4, l16 = lane & 15;
  constexpr int NRT = RPB / 16;
  int rt = wave % NRT;
  const float* xp = &Xs[(rt * 16 + l16) * DIN + 2 * h];
  for (int ct = wave / NRT; ct < DOUT / 16; ct += 4 / NRT) {
    int n0 = ct * 16;
    const float* wpp = &Ws[2 * h * DOUT + n0 + l16];
    v8f c = {};
    #pragma unroll
    for (int s = 0; s < DIN / 4; ++s) {
      v2f a; a.x = xp[4 * s];           a.y = xp[4 * s + 1];
      v2f b; b.x = wpp[4 * s * DOUT];   b.y = wpp[(4 * s + 1) * DOUT];
      c = __builtin_amdgcn_wmma_f32_16x16x4_f32(false, a, false, b, (short)0, c,
                                                false, false);
    }
    #pragma unroll
    for (int r = 0; r < 8; ++r) {
      int i = base + rt * 16 + r + 8 * h;
      if (i < NUM_ENT)
        acc[(size_t)i * DOUT + n0 + l16] = c[r];     // first writer, plain store
    }
  }
}

// ------------------------------- epilogue -----------------------------------

__global__ void colstat_kernel(const float* __restrict__ acc,
                               float* __restrict__ colsum, float* __restrict__ colsq,
                               int N, int D) {
  int o = threadIdx.x;
  float s = 0.f, q = 0.f;
  for (int i = blockIdx.x; i < N; i += gridDim.x) {
    float v = acc[(size_t)i * D + o];
    s += v; q += v * v;
  }
  atomicAdd(&colsum[o], s);
  atomicAdd(&colsq[o], q);
}

// out_pre = acc/3 + bias; BN mean-subtraction cancels bias.
__global__ void finalize_kernel(const float* __restrict__ acc,
                                const float* __restrict__ colsum,
                                const float* __restrict__ colsq,
                                const float* __restrict__ gamma,
                                const float* __restrict__ beta,
                                float* __restrict__ xout, int N, int D) {
  for (long idx = (long)blockIdx.x * blockDim.x + threadIdx.x; idx < (long)N * D;
       idx += (long)gridDim.x * blockDim.x) {
    int o = (int)(idx % D);
    float mean = colsum[o] / (float)N;
    float var  = colsq[o] / (float)N - mean * mean;
    float rstd = rsqrtf(var * (1.0f / 9.0f) + 1e-5f);
    float y = (acc[idx] - mean) * (1.0f / 3.0f) * rstd * gamma[o] + beta[o];
    xout[idx] = tanhf(y);
  }
}

__global__ void relmm_kernel(const float* __restrict__ rel, const float* __restrict__ wrel,
                             float* __restrict__ rel2, int DIN, int DOUT) {
  int t = blockIdx.x, o = threadIdx.x;
  float s = 0.f;
  for (int k = 0; k < DIN; ++k) s += rel[t * DIN + k] * wrel[k * DOUT + o];
  rel2[t * DOUT + o] = s;
}

// --------------------------------- driver -----------------------------------

extern "C" void kernel_launch(void* const* d_in, const int* in_sizes, int n_in,
                              void* d_out, int out_size, void* d_ws, size_t ws_size,
                              hipStream_t stream) {
  (void)in_sizes; (void)n_in; (void)out_size;
  const int*   ei         = (const int*)  d_in[0];
  const int*   etype      = (const int*)  d_in[1];
  const float* init_emb   = (const float*)d_in[2];
  const float* init_rel   = (const float*)d_in[3];
  const float* c1_w_in    = (const float*)d_in[4];
  const float* c1_w_out   = (const float*)d_in[5];
  const float* c1_w_loop  = (const float*)d_in[6];
  const float* c1_w_rel   = (const float*)d_in[7];
  const float* c1_looprel = (const float*)d_in[8];
  const float* c1_gamma   = (const float*)d_in[10];
  const float* c1_beta    = (const float*)d_in[11];
  const float* c2_w_in    = (const float*)d_in[12];
  const float* c2_w_out   = (const float*)d_in[13];
  const float* c2_w_loop  = (const float*)d_in[14];
  const float* c2_looprel = (const float*)d_in[16];
  const float* c2_gamma   = (const float*)d_in[18];
  const float* c2_beta    = (const float*)d_in[19];
  float* out = (float*)d_out;

  // -------- workspace carve (256B aligned regions) --------
  size_t off = 0;
  auto carve = [&](size_t bytes) {
    size_t o = off;
    off = (off + bytes + 255) & ~(size_t)255;
    return (char*)d_ws + o;
  };
  float* dinv_in   = (float*)carve(NUM_ENT * 4);
  float* dinv_out  = (float*)carve(NUM_ENT * 4);
  int*   cnt       = (int*)  carve(256 * 4);
  int*   seg_fill  = (int*)  carve(256 * 4);
  int*   segcs     = (int*)  carve(2 * 101 * 4);
  int*   ctype     = (int*)  carve(2 * NCHUNK * 4);
  int*   sorted    = (int*)  carve((size_t)2 * NCHPAD * 4);
  float* Wp_in     = (float*)carve((size_t)NREL2 * 128 * 64 * 4);  // din*dout=8192/type
  float* Wp_out    = (float*)carve((size_t)NREL2 * 128 * 64 * 4);
  float* Wp_loop   = (float*)carve(128 * 64 * 4);
  float* acc       = (float*)carve((size_t)NUM_ENT * 128 * 4);
  float* x1        = (float*)carve((size_t)NUM_ENT * 128 * 4);
  float* rel2      = (float*)carve(NREL2 * 128 * 4);
  float* colsum    = (float*)carve(256 * 4);        // [128 sum | 128 sumsq]
  float* colsq     = colsum + 128;
  if (off > ws_size) return;                        // insufficient scratch

  const int* rows_in  = ei;
  const int* cols_in  = ei + E_TOT;
  const int* rows_out = ei + E_HALF;
  const int* cols_out = ei + E_TOT + E_HALF;
  int* sorted_in  = sorted;
  int* sorted_out = sorted + NCHPAD;
  int* ctype_in   = ctype;
  int* ctype_out  = ctype + NCHUNK;

  // -------- structural preprocessing (shared by both layers) --------
  hipMemsetAsync(dinv_in,  0,    NUM_ENT * 4, stream);
  hipMemsetAsync(dinv_out, 0,    NUM_ENT * 4, stream);
  hipMemsetAsync(cnt,      0,    256 * 4, stream);
  hipMemsetAsync(sorted,   0xFF, (size_t)2 * NCHPAD * 4, stream);
  deg_kernel <<<2048, 256, 0, stream>>>(ei, dinv_in, dinv_out);
  dinv_kernel<<<(NUM_ENT + 255) / 256, 256, 0, stream>>>(dinv_in, dinv_out);
  hist_kernel<<<1024, 256, 0, stream>>>(etype, cnt);
  plan_kernel<<<1, 1, 0, stream>>>(cnt, seg_fill, segcs);
  chunkfill_kernel<<<(2 * NCHUNK + 255) / 256, 256, 0, stream>>>(segcs, ctype);
  scatter_kernel<<<2048, 256, 0, stream>>>(etype, seg_fill, sorted);

  // -------- layer 1: 64 -> 128 (64 edges/block, 48.75 KB LDS) --------
  wprime_kernel<<<dim3(64, NREL2), 128, 0, stream>>>(init_rel,   c1_w_in,   Wp_in,   64, 128);
  wprime_kernel<<<dim3(64, NREL2), 128, 0, stream>>>(init_rel,   c1_w_out,  Wp_out,  64, 128);
  wprime_kernel<<<dim3(64, 1),     128, 0, stream>>>(c1_looprel, c1_w_loop, Wp_loop, 64, 128);
  loop_wmma_kernel<64, 128, 64><<<(NUM_ENT + 63) / 64, 128, 0, stream>>>(init_emb, Wp_loop, acc);
  edge_wmma_kernel<64, 128, 64><<<NCHUNK, 128, 0, stream>>>(
      init_emb, Wp_in,  sorted_in,  ctype_in,  rows_in,  cols_in,  dinv_in,  acc);
  edge_wmma_kernel<64, 128, 64><<<NCHUNK, 128, 0, stream>>>(
      init_emb, Wp_out, sorted_out, ctype_out, rows_out, cols_out, dinv_out, acc);
  hipMemsetAsync(colsum, 0, 256 * 4, stream);
  colstat_kernel <<<512, 128, 0, stream>>>(acc, colsum, colsq, NUM_ENT, 128);
  finalize_kernel<<<50000, 256, 0, stream>>>(acc, colsum, colsq, c1_gamma, c1_beta,
                                             x1, NUM_ENT, 128);
  relmm_kernel<<<NREL2, 128, 0, stream>>>(init_rel, c1_w_rel, rel2, 64, 128);

  // -------- layer 2: 128 -> 64 (32 edges/block, 48.375 KB LDS) --------
  wprime_kernel<<<dim3(128, NREL2), 64, 0, stream>>>(rel2,       c2_w_in,   Wp_in,   128, 64);
  wprime_kernel<<<dim3(128, NREL2), 64, 0, stream>>>(rel2,       c2_w_out,  Wp_out,  128, 64);
  wprime_kernel<<<dim3(128, 1),     64, 0, stream>>>(c2_looprel, c2_w_loop, Wp_loop, 128, 64);
  loop_wmma_kernel<128, 64, 32><<<NUM_ENT / 32, 128, 0, stream>>>(x1, Wp_loop, acc);
  edge_wmma_kernel<128, 64, 32><<<2 * NCHUNK, 128, 0, stream>>>(
      x1, Wp_in,  sorted_in,  ctype_in,  rows_in,  cols_in,  dinv_in,  acc);
  edge_wmma_kernel<128, 64, 32><<<2 * NCHUNK, 128, 0, stream>>>(
      x1, Wp_out, sorted_out, ctype_out, rows_out, cols_out, dinv_out, acc);
  hipMemsetAsync(colsum, 0, 256 * 4, stream);
  colstat_kernel <<<512, 64, 0, stream>>>(acc, colsum, colsq, NUM_ENT, 64);
  finalize_kernel<<<25000, 256, 0, stream>>>(acc, colsum, colsq, c2_gamma, c2_beta,
                                             out, NUM_ENT, 64);
}